// GuidedFilter_309237645450
// MI455X (gfx1250) — compile-verified
//
#include <hip/hip_runtime.h>

// Guided filter, r=40, on 24 planes of 512x512 f32.
// Box sums are computed as banded-matrix matmuls on the f32 WMMA pipe
// (V_WMMA_F32_16X16X4_F32), keeping full f32 precision for the
// cancellation-sensitive var/cov terms. Intermediates live in d_ws and are
// L2-resident (<=144MB < 192MB L2), so HBM traffic is ~I/p reads + final write.

#define RW   40
#define EPSF 0.001f
#define W    512
#define PSZ  (512 * 512)
#define NPLANES 24

typedef __attribute__((ext_vector_type(2))) float v2f;
typedef __attribute__((ext_vector_type(8))) float v8f;

__device__ __forceinline__ v8f wmma4(v2f a, v2f b, v8f c) {
  // D = A(16x4) * B(4x16) + C, all f32. 8-arg form per CDNA5 probe notes.
  return __builtin_amdgcn_wmma_f32_16x16x4_f32(
      /*neg_a=*/false, a, /*neg_b=*/false, b,
      /*c_mod=*/(short)0, c, /*reuse_a=*/false, /*reuse_b=*/false);
}

__device__ __forceinline__ int   imin(int a, int b) { return a < b ? a : b; }
__device__ __forceinline__ int   imax(int a, int b) { return a > b ? a : b; }
// 1.0 if |d| <= RW else 0.0
__device__ __forceinline__ float bandv(int d) {
  return ((unsigned)(d + RW) <= (unsigned)(2 * RW)) ? 1.0f : 0.0f;
}
// window pixel count along one axis at coordinate t (zero-padded box)
__device__ __forceinline__ float cntf(int t) {
  return (float)(imin(t + RW, W - 1) - imax(t - RW, 0) + 1);
}

// ---------------- Stage 1: horizontal box sums of I, p, I*p, I*I ------------
__global__ __launch_bounds__(256) void gf_hpass4(
    const float* __restrict__ I, const float* __restrict__ p,
    float* __restrict__ sI, float* __restrict__ sP,
    float* __restrict__ sIp, float* __restrict__ sII) {
  const int lane = threadIdx.x & 31;
  const int lo = lane & 15, hi = lane >> 4;
  const int tile = blockIdx.x * 8 + (threadIdx.x >> 5);
  const int y0 = (tile >> 5) * 16, x0 = (tile & 31) * 16;
  const size_t pb = (size_t)blockIdx.y * PSZ;

  // A operand = image data: lane holds row y0+lo, cols (u0+2*hi, u0+2*hi+1)
  const float* Ir = I + pb + (size_t)(y0 + lo) * W;
  const float* pr = p + pb + (size_t)(y0 + lo) * W;

  v8f aI = {}, aP = {}, aIp = {}, aII = {};
  const int x = x0 + lo;                      // B operand column (N = lo)
  const int uS = imax(0, x0 - RW) & ~3;
  const int uE = imin(W, x0 + 16 + RW);
  for (int u0 = uS; u0 < uE; u0 += 4) {
    const int ac = u0 + 2 * hi;               // even => b64-aligned pair
    v2f vi = {0.f, 0.f}, vp = {0.f, 0.f};
    if (ac < W) {
      vi = *reinterpret_cast<const v2f*>(Ir + ac);
      vp = *reinterpret_cast<const v2f*>(pr + ac);
    }
    const v2f vip = vi * vp;
    const v2f vii = vi * vi;
    const v2f bb = {bandv(ac - x), bandv(ac + 1 - x)};  // banded 0/1 matrix
    aI  = wmma4(vi,  bb, aI);
    aP  = wmma4(vp,  bb, aP);
    aIp = wmma4(vip, bb, aIp);
    aII = wmma4(vii, bb, aII);
  }

  const size_t ob = pb + (size_t)(y0 + 8 * hi) * W + (x0 + lo);
#pragma unroll
  for (int g = 0; g < 8; ++g) {
    sI [ob + (size_t)g * W] = aI[g];
    sP [ob + (size_t)g * W] = aP[g];
    sIp[ob + (size_t)g * W] = aIp[g];
    sII[ob + (size_t)g * W] = aII[g];
  }
}

// ------- Stage 2: vertical box sums + per-pixel a = cov/(var+eps), b -------
__global__ __launch_bounds__(256) void gf_vpass4_ab(
    const float* __restrict__ sI, const float* __restrict__ sP,
    const float* __restrict__ sIp, const float* __restrict__ sII,
    float* __restrict__ aOut, float* __restrict__ bOut) {
  const int lane = threadIdx.x & 31;
  const int lo = lane & 15, hi = lane >> 4;
  const int tile = blockIdx.x * 8 + (threadIdx.x >> 5);
  const int y0 = (tile >> 5) * 16, x0 = (tile & 31) * 16;
  const size_t pb = (size_t)blockIdx.y * PSZ;

  const int col = x0 + lo;      // B operand column (N = lo)
  const int ym  = y0 + lo;      // A operand row (M = lo)

  v8f cI = {}, cP = {}, cIp = {}, cII = {};
  const int vS = imax(0, y0 - RW) & ~3;
  const int vE = imin(W, y0 + 16 + RW);
  for (int v0 = vS; v0 < vE; v0 += 4) {
    const int r = v0 + 2 * hi;
    const v2f ab = {bandv(r - ym), bandv(r + 1 - ym)};   // band as A
    v2f bI = {0.f, 0.f}, bP = {0.f, 0.f}, bIp = {0.f, 0.f}, bII = {0.f, 0.f};
    const size_t base = pb + (size_t)r * W + col;
    if (r < W) {
      bI.x = sI[base]; bP.x = sP[base]; bIp.x = sIp[base]; bII.x = sII[base];
    }
    if (r + 1 < W) {
      bI.y = sI[base + W]; bP.y = sP[base + W];
      bIp.y = sIp[base + W]; bII.y = sII[base + W];
    }
    cI  = wmma4(ab, bI,  cI);
    cP  = wmma4(ab, bP,  cP);
    cIp = wmma4(ab, bIp, cIp);
    cII = wmma4(ab, bII, cII);
  }

  const float cx = cntf(col);
  const size_t ob = pb + (size_t)(y0 + 8 * hi) * W + col;
#pragma unroll
  for (int g = 0; g < 8; ++g) {
    const int y = y0 + 8 * hi + g;
    const float invN = 1.0f / (cntf(y) * cx);
    const float mI = cI[g] * invN, mP = cP[g] * invN;
    const float mIp = cIp[g] * invN, mII = cII[g] * invN;
    const float varI = mII - mI * mI;
    const float cov  = mIp - mI * mP;
    const float av = cov / (varI + EPSF);
    const float bv = mP - av * mI;
    aOut[ob + (size_t)g * W] = av;
    bOut[ob + (size_t)g * W] = bv;
  }
}

// ---------------- Stage 3: horizontal box sums of a, b ----------------------
__global__ __launch_bounds__(256) void gf_hpass2(
    const float* __restrict__ aIn, const float* __restrict__ bIn,
    float* __restrict__ ah, float* __restrict__ bh) {
  const int lane = threadIdx.x & 31;
  const int lo = lane & 15, hi = lane >> 4;
  const int tile = blockIdx.x * 8 + (threadIdx.x >> 5);
  const int y0 = (tile >> 5) * 16, x0 = (tile & 31) * 16;
  const size_t pb = (size_t)blockIdx.y * PSZ;

  const float* ar = aIn + pb + (size_t)(y0 + lo) * W;
  const float* br = bIn + pb + (size_t)(y0 + lo) * W;

  v8f ca = {}, cb = {};
  const int x = x0 + lo;
  const int uS = imax(0, x0 - RW) & ~3;
  const int uE = imin(W, x0 + 16 + RW);
  for (int u0 = uS; u0 < uE; u0 += 4) {
    const int ac = u0 + 2 * hi;
    v2f va = {0.f, 0.f}, vb = {0.f, 0.f};
    if (ac < W) {
      va = *reinterpret_cast<const v2f*>(ar + ac);
      vb = *reinterpret_cast<const v2f*>(br + ac);
    }
    const v2f bb = {bandv(ac - x), bandv(ac + 1 - x)};
    ca = wmma4(va, bb, ca);
    cb = wmma4(vb, bb, cb);
  }

  const size_t ob = pb + (size_t)(y0 + 8 * hi) * W + (x0 + lo);
#pragma unroll
  for (int g = 0; g < 8; ++g) {
    ah[ob + (size_t)g * W] = ca[g];
    bh[ob + (size_t)g * W] = cb[g];
  }
}

// ----- Stage 4: vertical box sums of a, b; out = mean_a * I + mean_b --------
__global__ __launch_bounds__(256) void gf_vpass2_final(
    const float* __restrict__ ah, const float* __restrict__ bh,
    const float* __restrict__ I, float* __restrict__ out) {
  const int lane = threadIdx.x & 31;
  const int lo = lane & 15, hi = lane >> 4;
  const int tile = blockIdx.x * 8 + (threadIdx.x >> 5);
  const int y0 = (tile >> 5) * 16, x0 = (tile & 31) * 16;
  const size_t pb = (size_t)blockIdx.y * PSZ;

  const int col = x0 + lo;
  const int ym  = y0 + lo;

  v8f ca = {}, cb = {};
  const int vS = imax(0, y0 - RW) & ~3;
  const int vE = imin(W, y0 + 16 + RW);
  for (int v0 = vS; v0 < vE; v0 += 4) {
    const int r = v0 + 2 * hi;
    const v2f ab = {bandv(r - ym), bandv(r + 1 - ym)};
    v2f ba = {0.f, 0.f}, bb = {0.f, 0.f};
    const size_t base = pb + (size_t)r * W + col;
    if (r < W)     { ba.x = ah[base];     bb.x = bh[base]; }
    if (r + 1 < W) { ba.y = ah[base + W]; bb.y = bh[base + W]; }
    ca = wmma4(ab, ba, ca);
    cb = wmma4(ab, bb, cb);
  }

  const float cx = cntf(col);
  const size_t ob = pb + (size_t)(y0 + 8 * hi) * W + col;
#pragma unroll
  for (int g = 0; g < 8; ++g) {
    const int y = y0 + 8 * hi + g;
    const float invN = 1.0f / (cntf(y) * cx);
    const float meanA = ca[g] * invN;
    const float meanB = cb[g] * invN;
    const size_t idx = ob + (size_t)g * W;
    out[idx] = meanA * I[idx] + meanB;
  }
}

extern "C" void kernel_launch(void* const* d_in, const int* in_sizes, int n_in,
                              void* d_out, int out_size, void* d_ws, size_t ws_size,
                              hipStream_t stream) {
  (void)in_sizes; (void)n_in; (void)out_size; (void)ws_size;
  const float* I = (const float*)d_in[0];
  const float* p = (const float*)d_in[1];
  float* ws = (float*)d_ws;

  // Scratch layout (24 planes each, 1MB/plane): peak 144MB, fits 192MB L2.
  float* sI  = ws;
  float* sP  = ws + (size_t)24 * PSZ;
  float* sIp = ws + (size_t)48 * PSZ;
  float* sII = ws + (size_t)72 * PSZ;
  float* aB  = ws + (size_t)96 * PSZ;
  float* bB  = ws + (size_t)120 * PSZ;
  float* ah  = sI;   // sI/sP regions are dead after stage 2 -> reuse
  float* bh  = sP;

  dim3 block(256);            // 8 waves; each wave owns one 16x16 tile
  dim3 grid(128, NPLANES);    // 128 * 8 = 1024 tiles = (512/16)^2 per plane

  gf_hpass4      <<<grid, block, 0, stream>>>(I, p, sI, sP, sIp, sII);
  gf_vpass4_ab   <<<grid, block, 0, stream>>>(sI, sP, sIp, sII, aB, bB);
  gf_hpass2      <<<grid, block, 0, stream>>>(aB, bB, ah, bh);
  gf_vpass2_final<<<grid, block, 0, stream>>>(ah, bh, I, (float*)d_out);
}